// DenoisingAutoencoder_71330816852320
// MI455X (gfx1250) — compile-verified
//
#include <hip/hip_runtime.h>
#include <math.h>

// ---------------------------------------------------------------------------
// 2-layer GAT for MI455X (gfx1250, wave32).
// GEMMs use V_WMMA_F32_16X16X4_F32 (full fp32 precision, matrix pipe).
// Edge phase: compacted int32 edge list, float atomic max/sum, float4 scatter.
// ---------------------------------------------------------------------------

typedef __attribute__((ext_vector_type(2))) float v2f;
typedef __attribute__((ext_vector_type(8))) float v8f;

#define NEG_SLOPE 0.2f

__device__ __forceinline__ void atomicMaxF(float* addr, float v) {
  // fp32-exact atomic max via sign-split trick; addr must be seeded with -inf.
  if (v >= 0.0f) atomicMax((int*)addr, __float_as_int(v));
  else           atomicMin((unsigned int*)addr, __float_as_uint(v));
}

// ---- build compact int32 edge list (with self loops appended) --------------
__global__ void k_build_eidx(const long long* __restrict__ ei,
                             int2* __restrict__ eidx, int E, int N) {
  int e = blockIdx.x * blockDim.x + threadIdx.x;
  int ET = E + N;
  if (e >= ET) return;
  int2 sd;
  if (e < E) { sd.x = (int)ei[e]; sd.y = (int)ei[(long long)E + e]; }
  else       { sd.x = sd.y = e - E; }
  eidx[e] = sd;
}

__global__ void k_fill_f32(float* __restrict__ p, float v, int n) {
  int i = blockIdx.x * blockDim.x + threadIdx.x;
  if (i < n) p[i] = v;
}

// ---- C[M x Nout] = A[M x K] @ W[K x Nout] via V_WMMA_F32_16X16X4_F32 -------
// One wave computes one 16x16 tile of C; K-loop steps by 4.
__global__ void k_gemm_wmma(const float* __restrict__ A,
                            const float* __restrict__ W,
                            float* __restrict__ C,
                            int M, int K, int Nout) {
  int lane = threadIdx.x & 31;
  int wave = (blockIdx.x * blockDim.x + threadIdx.x) >> 5;
  int tilesN = Nout >> 4;
  int tilesM = M >> 4;
  if (wave >= tilesM * tilesN) return;   // wave-uniform: EXEC stays all-ones
  int tm = wave / tilesN;
  int tn = wave - tm * tilesN;

  int mi = lane & 15;            // row (A) / col (B) within tile
  int kb = (lane >> 4) << 1;     // half-wave K pair: 0 or 2

  const float* __restrict__ Arow = A + (long long)(tm * 16 + mi) * K;
  const float* __restrict__ Wcol = W + tn * 16 + mi;

  v8f acc = {};
  for (int kk = 0; kk < K; kk += 4) {
    v2f a, b;
    a.x = Arow[kk + kb];
    a.y = Arow[kk + kb + 1];
    b.x = Wcol[(long long)(kk + kb) * Nout];
    b.y = Wcol[(long long)(kk + kb + 1) * Nout];
    acc = __builtin_amdgcn_wmma_f32_16x16x4_f32(
        /*neg_a=*/false, a, /*neg_b=*/false, b,
        /*c_mod=*/(short)0, acc, /*reuse_a=*/false, /*reuse_b=*/false);
  }

  // C/D layout: VGPR v, lane l -> row v + 8*(l>=16), col l&15
  int row0 = tm * 16 + ((lane >> 4) << 3);
  float* __restrict__ Cout = C + (long long)row0 * Nout + tn * 16 + (lane & 15);
#pragma unroll
  for (int v = 0; v < 8; ++v) Cout[(long long)v * Nout] = acc[v];
}

// ---- per-node attention scores: s[n]=h[n].a_s, d[n]=h[n].a_d ---------------
__global__ void k_node_scores(const float* __restrict__ h,
                              const float* __restrict__ as,
                              const float* __restrict__ ad,
                              float* __restrict__ s, float* __restrict__ d,
                              int N, int F) {
  int n = blockIdx.x * blockDim.x + threadIdx.x;
  if (n >= N) return;
  const float* row = h + (long long)n * F;
  float ss = 0.0f, dd = 0.0f;
  for (int f = 0; f < F; f += 4) {
    float4 hv = *(const float4*)(row + f);
    float4 av = *(const float4*)(as + f);
    float4 bv = *(const float4*)(ad + f);
    ss += hv.x * av.x + hv.y * av.y + hv.z * av.z + hv.w * av.w;
    dd += hv.x * bv.x + hv.y * bv.y + hv.z * bv.z + hv.w * bv.w;
  }
  s[n] = ss;
  d[n] = dd;
}

// ---- per-edge leaky-relu logit + segment max over dst ----------------------
__global__ void k_edge_logit_max(const int2* __restrict__ eidx,
                                 const float* __restrict__ s,
                                 const float* __restrict__ d,
                                 float* __restrict__ elog,
                                 float* __restrict__ mx, int ET) {
  int e = blockIdx.x * blockDim.x + threadIdx.x;
  if (e >= ET) return;
  int2 sd = eidx[e];
  float l = s[sd.x] + d[sd.y];
  l = (l > 0.0f) ? l : NEG_SLOPE * l;
  elog[e] = l;
  atomicMaxF(&mx[sd.y], l);
}

// ---- e = exp(logit - max[dst]); denom[dst] += e ----------------------------
__global__ void k_edge_exp_sum(const int2* __restrict__ eidx,
                               float* __restrict__ elog,
                               const float* __restrict__ mx,
                               float* __restrict__ denom, int ET) {
  int e = blockIdx.x * blockDim.x + threadIdx.x;
  if (e >= ET) return;
  int2 sd = eidx[e];
  float ex = __expf(elog[e] - mx[sd.y]);
  elog[e] = ex;
  atomicAdd(&denom[sd.y], ex);
}

// ---- out[dst] += (e/denom[dst]) * h[src]; (1<<tpeShift) lanes per edge -----
__global__ void k_edge_scatter(const int2* __restrict__ eidx,
                               const float* __restrict__ eexp,
                               const float* __restrict__ denom,
                               const float* __restrict__ h,
                               float* __restrict__ out,
                               int ET, int F, int tpeShift) {
  long long t = (long long)blockIdx.x * blockDim.x + threadIdx.x;
  long long e = t >> tpeShift;
  if (e >= ET) return;
  int fq = ((int)t & ((1 << tpeShift) - 1)) << 2;
  int2 sd = eidx[e];
  float coef = eexp[e] / (denom[sd.y] + 1e-16f);
  float4 hv = *(const float4*)(h + (long long)sd.x * F + fq);
  float* o = out + (long long)sd.y * F + fq;
  atomicAdd(o + 0, coef * hv.x);
  atomicAdd(o + 1, coef * hv.y);
  atomicAdd(o + 2, coef * hv.z);
  atomicAdd(o + 3, coef * hv.w);
}

__global__ void k_bias_relu(float* __restrict__ p, const float* __restrict__ b,
                            int total, int fmask) {
  int i = blockIdx.x * blockDim.x + threadIdx.x;
  if (i >= total) return;
  float v = p[i] + b[i & fmask];
  p[i] = (v > 0.0f) ? v : 0.0f;
}

__global__ void k_bias_add(float* __restrict__ p, const float* __restrict__ b,
                           int total, int fmask) {
  int i = blockIdx.x * blockDim.x + threadIdx.x;
  if (i >= total) return;
  p[i] = p[i] + b[i & fmask];
}

// ---------------------------------------------------------------------------
extern "C" void kernel_launch(void* const* d_in, const int* in_sizes, int n_in,
                              void* d_out, int out_size, void* d_ws, size_t ws_size,
                              hipStream_t stream) {
  const float*     x   = (const float*)d_in[0];
  const long long* ei  = (const long long*)d_in[1];  // int64 edge_index [2,E]
  const float*     W1  = (const float*)d_in[2];
  const float*     as1 = (const float*)d_in[3];
  const float*     ad1 = (const float*)d_in[4];
  const float*     b1  = (const float*)d_in[5];
  const float*     W2  = (const float*)d_in[6];
  const float*     as2 = (const float*)d_in[7];
  const float*     ad2 = (const float*)d_in[8];
  const float*     b2  = (const float*)d_in[9];

  const int N  = in_sizes[0] / 64;   // 80000
  const int E  = in_sizes[1] / 2;    // 1280000
  const int ET = E + N;              // + self loops
  float* out = (float*)d_out;

  // workspace carve-up (~48 MB total)
  char* w = (char*)d_ws;
  int2*  eidx  = (int2*)w;   w += (size_t)ET * sizeof(int2);
  float* h     = (float*)w;  w += (size_t)N * 64 * sizeof(float);  // h1 (N*32) / h2 (N*64)
  float* out1  = (float*)w;  w += (size_t)N * 32 * sizeof(float);
  float* sbuf  = (float*)w;  w += (size_t)N * sizeof(float);
  float* dbuf  = (float*)w;  w += (size_t)N * sizeof(float);
  float* mx    = (float*)w;  w += (size_t)N * sizeof(float);
  float* denom = (float*)w;  w += (size_t)N * sizeof(float);
  float* elog  = (float*)w;  w += (size_t)ET * sizeof(float);

  const int B = 256;
  auto cdiv = [](long long a, long long b) { return (int)((a + b - 1) / b); };

  k_build_eidx<<<cdiv(ET, B), B, 0, stream>>>(ei, eidx, E, N);
  hipMemsetAsync(out1, 0, (size_t)N * 32 * sizeof(float), stream);
  hipMemsetAsync(out,  0, (size_t)N * 64 * sizeof(float), stream);

  // ---------------- layer 1: GATConv(64 -> 32) ----------------
  {
    long long waves = (long long)(N / 16) * (32 / 16);
    k_gemm_wmma<<<cdiv(waves * 32, B), B, 0, stream>>>(x, W1, h, N, 64, 32);
    k_node_scores<<<cdiv(N, B), B, 0, stream>>>(h, as1, ad1, sbuf, dbuf, N, 32);
    k_fill_f32<<<cdiv(N, B), B, 0, stream>>>(mx, -INFINITY, N);
    hipMemsetAsync(denom, 0, (size_t)N * sizeof(float), stream);
    k_edge_logit_max<<<cdiv(ET, B), B, 0, stream>>>(eidx, sbuf, dbuf, elog, mx, ET);
    k_edge_exp_sum<<<cdiv(ET, B), B, 0, stream>>>(eidx, elog, mx, denom, ET);
    k_edge_scatter<<<cdiv((long long)ET * 8, B), B, 0, stream>>>(
        eidx, elog, denom, h, out1, ET, 32, 3);
    k_bias_relu<<<cdiv((long long)N * 32, B), B, 0, stream>>>(out1, b1, N * 32, 31);
  }

  // ---------------- layer 2: GATConv(32 -> 64) ----------------
  {
    long long waves = (long long)(N / 16) * (64 / 16);
    k_gemm_wmma<<<cdiv(waves * 32, B), B, 0, stream>>>(out1, W2, h, N, 32, 64);
    k_node_scores<<<cdiv(N, B), B, 0, stream>>>(h, as2, ad2, sbuf, dbuf, N, 64);
    k_fill_f32<<<cdiv(N, B), B, 0, stream>>>(mx, -INFINITY, N);
    hipMemsetAsync(denom, 0, (size_t)N * sizeof(float), stream);
    k_edge_logit_max<<<cdiv(ET, B), B, 0, stream>>>(eidx, sbuf, dbuf, elog, mx, ET);
    k_edge_exp_sum<<<cdiv(ET, B), B, 0, stream>>>(eidx, elog, mx, denom, ET);
    k_edge_scatter<<<cdiv((long long)ET * 16, B), B, 0, stream>>>(
        eidx, elog, denom, h, out, ET, 64, 4);
    k_bias_add<<<cdiv((long long)N * 64, B), B, 0, stream>>>(out, b2, N * 64, 63);
  }
}